// CustomConvLayer_55413668053303
// MI455X (gfx1250) — compile-verified
//
#include <hip/hip_runtime.h>

typedef __attribute__((ext_vector_type(2))) float v2f;
typedef __attribute__((ext_vector_type(8))) float v8f;

#define B_   16
#define C_   64
#define H_   128
#define W_   128
#define O_   128
#define KT   9
#define KGEMM (C_ * KT)        // 576
#define PXT   64               // pixels (W span) per workgroup
#define XCOLS (PXT + 2)        // 66 staged columns (halo of 1 each side)
#define CPAIRS (C_ / 2)        // 32 channel pairs
#define P_ROW (3 * XCOLS)      // float2 stride per channel-pair (198)

// Implicit-GEMM 3x3 conv, tap-major gemm-K: k = t*64 + c.
// LDS stores channel-PAIR-interleaved float2 {x[c], x[c+1]} so each WMMA
// B fragment (pair k0,k1 = c,c+1) is a single aligned ds_load_b64.
__global__ __launch_bounds__(256)
void conv3x3_wmma_f32(const float* __restrict__ x,
                      const float* __restrict__ wgt,
                      float* __restrict__ out)
{
    __shared__ v2f xs2[CPAIRS * 3 * XCOLS];   // 6336 float2 = 50,688 bytes

    const int tid = threadIdx.x;
    const int wt  = blockIdx.x & 1;           // which 64-px half of the row
    const int h   = (blockIdx.x >> 1) & (H_ - 1);
    const int b   = blockIdx.x >> 8;          // 2*128 blocks per batch image
    const int w0  = wt * PXT;

    // ---- stage x slab into LDS, channel-pair interleaved, zero halo ----
    float* xsf = (float*)xs2;
    for (int i = tid; i < C_ * 3 * XCOLS; i += 256) {
        const int col = i % XCOLS;
        const int rc  = i / XCOLS;
        const int r   = rc % 3;
        const int c   = rc / 3;
        const int gh  = h + r - 1;
        const int gw  = w0 + col - 1;
        float v = 0.0f;
        if ((unsigned)gh < (unsigned)H_ && (unsigned)gw < (unsigned)W_)
            v = x[((b * C_ + c) * H_ + gh) * W_ + gw];
        // float2-interleaved: element [(c/2)*3 + r][col], lane (c&1)
        xsf[((((c >> 1) * 3) + r) * XCOLS + col) * 2 + (c & 1)] = v;
    }
    __syncthreads();

    const int lane  = tid & 31;
    const int lr    = lane & 15;        // row of A / col of B / col of D
    const int half  = lane >> 4;        // selects K pair (A,B) / +8 rows (D)
    const int obase = (tid >> 5) * 16;  // 8 waves -> 8 o-tiles

    v8f acc0 = {}, acc1 = {}, acc2 = {}, acc3 = {};

    // A element for gemm-k = t*64+c is wgt flat [o*576 + c*9 + t].
    // This lane's K pair starts at c = 2*half  -> channel pair index = half.
    const float* wbase  = wgt + (obase + lr) * KGEMM + half * (2 * KT);
    const v2f*   x2base = xs2 + half * P_ROW + lr;

    for (int t = 0; t < KT; ++t) {           // uniform scalar tap loop
        const int dh = t / 3;
        const int dw = t - 3 * dh;
        const float* arow = wbase + t;                 // advance by tap
        const v2f*   brow = x2base + dh * XCOLS + dw;  // tap halo offset

        #pragma unroll 4
        for (int cb = 0; cb < C_; cb += 4) {
            // A fragment: a[j] = wgt[o = obase+lr][c = cb+2*half+j][t]
            v2f a;
            a.x = arow[cb * KT];
            a.y = arow[cb * KT + KT];

            // B fragments: one b64 load each = {B[k0][n], B[k1][n]}
            const v2f* bp = brow + (cb >> 1) * P_ROW;
            const v2f b0 = bp[0];
            const v2f b1 = bp[16];
            const v2f b2 = bp[32];
            const v2f b3 = bp[48];

            // 4 independent accumulator chains -> XDL pipelining
            acc0 = __builtin_amdgcn_wmma_f32_16x16x4_f32(false, a, false, b0,
                                                         (short)0, acc0, false, false);
            acc1 = __builtin_amdgcn_wmma_f32_16x16x4_f32(false, a, false, b1,
                                                         (short)0, acc1, false, false);
            acc2 = __builtin_amdgcn_wmma_f32_16x16x4_f32(false, a, false, b2,
                                                         (short)0, acc2, false, false);
            acc3 = __builtin_amdgcn_wmma_f32_16x16x4_f32(false, a, false, b3,
                                                         (short)0, acc3, false, false);
        }
    }

    // ---- store: D vgpr r holds row (obase + r + 8*half), col w0 + p*16 + lr
    float* po = out + ((b * O_ + obase + 8 * half) * H_ + h) * W_ + w0 + lr;
    #pragma unroll
    for (int r = 0; r < 8; ++r) {
        float* prow = po + r * (H_ * W_);
        prow[0]  = acc0[r];
        prow[16] = acc1[r];
        prow[32] = acc2[r];
        prow[48] = acc3[r];
    }
}

extern "C" void kernel_launch(void* const* d_in, const int* in_sizes, int n_in,
                              void* d_out, int out_size, void* d_ws, size_t ws_size,
                              hipStream_t stream) {
    (void)in_sizes; (void)n_in; (void)out_size; (void)d_ws; (void)ws_size;
    const float* x   = (const float*)d_in[0];
    const float* wgt = (const float*)d_in[1];
    float* out = (float*)d_out;

    const int grid = B_ * H_ * (W_ / PXT);   // 16 * 128 * 2 = 4096 workgroups
    conv3x3_wmma_f32<<<grid, 256, 0, stream>>>(x, wgt, out);
}